// MoEBlock_31181462569068
// MI455X (gfx1250) — compile-verified
//
#include <hip/hip_runtime.h>
#include <hip/hip_bf16.h>
#include <stdint.h>

// ---- Problem constants (match reference) ----
#define HIDDEN 1024
#define INTER  1408
#define NEXP   8
#define TOKENS 8192

// ---- Tiling ----
#define BM 64          // rows per block
#define BN 64          // cols per block
#define KC 16          // K slab per LDS stage
#define LDA 20         // padded LDS stride for A (floats)  == 16 data + 4 pad
#define LDB 68         // padded LDS stride for B (floats)  == 64 data + 4 pad

typedef float v2f __attribute__((ext_vector_type(2)));
typedef float v8f __attribute__((ext_vector_type(8)));
typedef unsigned int v4u __attribute__((ext_vector_type(4)));
typedef int v4i __attribute__((ext_vector_type(4)));
typedef int v8i __attribute__((ext_vector_type(8)));

#if __has_builtin(__builtin_amdgcn_tensor_load_to_lds)
#define USE_TDM 1
#if __has_include(<hip/amd_detail/amd_gfx1250_TDM.h>)
#define TDM_6ARG 1     // therock/clang-23 headers present -> 6-arg builtin
#else
#define TDM_6ARG 0     // ROCm 7.2 / clang-22 -> 5-arg builtin
#endif
#else
#define USE_TDM 0
#endif

__device__ __forceinline__ v8f wmma_f32(v2f a, v2f b, v8f c) {
    // D = A(16x4 f32) * B(4x16 f32) + C(16x16 f32)
    return __builtin_amdgcn_wmma_f32_16x16x4_f32(
        false, a, false, b, (short)0, c, false, false);
}

__device__ __forceinline__ int seg_start(const int* __restrict__ tpe, int e) {
    int s = 0;
    #pragma unroll
    for (int i = 0; i < NEXP; ++i) s += (i < e) ? tpe[i] : 0;
    return s;
}

#if USE_TDM
// Issue one 2D TDM tile load: global (row-major, stride_elems f32/row) -> LDS,
// with hardware row padding and zero-fill past [rem_w, rem_h] from tile origin.
// pad codes: interval 3 -> pad every 16 dwords, 5 -> every 64; amount 3 -> 4 dwords.
__device__ __forceinline__ void tdm_load_tile(
    void* lds_ptr, const void* gptr,
    unsigned tile_w, unsigned tile_h,
    unsigned rem_w,  unsigned rem_h,
    unsigned stride_elems,
    unsigned pad_interval_code, unsigned pad_amount_code)
{
    const unsigned long long ga = (unsigned long long)(uintptr_t)gptr;
    const unsigned lds_off = (unsigned)(uintptr_t)lds_ptr;   // generic addr low 32 = LDS byte offset
    v4u g0;
    g0[0] = 1u;                                              // count=1, user descriptor
    g0[1] = lds_off;                                         // lds_addr (bytes)
    g0[2] = (unsigned)ga;                                    // global_addr[31:0]
    g0[3] = (unsigned)((ga >> 32) & 0x01FFFFFFull) | (2u << 30); // addr[56:32] | type=2
    v8i g1;
    g1[0] = (int)((2u << 16) |                               // data_size = 4B
                  (1u << 20) |                               // pad_enable
                  (pad_interval_code << 22) |
                  (pad_amount_code   << 25));
    g1[1] = (int)((rem_w & 0xFFFFu) << 16);                  // tensor_dim0[15:0] @ bits63:48
    g1[2] = (int)(((rem_w >> 16) & 0xFFFFu) |                // tensor_dim0[31:16]
                  ((rem_h & 0xFFFFu) << 16));                // tensor_dim1[15:0]
    g1[3] = (int)(((rem_h >> 16) & 0xFFFFu) |                // tensor_dim1[31:16]
                  ((tile_w & 0xFFFFu) << 16));               // tile_dim0
    g1[4] = (int)(tile_h & 0xFFFFu);                         // tile_dim1 (tile_dim2 = 0)
    g1[5] = (int)stride_elems;                               // tensor_dim0_stride[31:0]
    g1[6] = 0;                                               // stride hi / dim1_stride lo
    g1[7] = 0;
    const v4i z4 = {0, 0, 0, 0};
#if TDM_6ARG
    const v8i z8 = {0, 0, 0, 0, 0, 0, 0, 0};
    __builtin_amdgcn_tensor_load_to_lds(g0, g1, z4, z4, z8, 0);
#else
    __builtin_amdgcn_tensor_load_to_lds(g0, g1, z4, z4, 0);
#endif
}
#endif // USE_TDM

// ======================= Kernel 1: gate/up GEMM + SwiGLU =======================
__global__ __launch_bounds__(256) void moe_gateup_swiglu_kernel(
    const float* __restrict__ x,       // [TOKENS, HIDDEN]
    const float* __restrict__ w1w3,    // [NEXP, HIDDEN, 2*INTER]
    const int*   __restrict__ tpe,     // [NEXP]
    float*       __restrict__ h)       // [TOKENS, INTER]
{
    __shared__ float As[2][BM * LDA];
    __shared__ float Bg[2][KC * LDB];
    __shared__ float Bu[2][KC * LDB];

    const int e      = blockIdx.z;
    const int segLen = __builtin_amdgcn_readfirstlane(tpe[e]);
    if ((int)blockIdx.y * BM >= segLen) return;
    const int segBeg  = __builtin_amdgcn_readfirstlane(seg_start(tpe, e));
    const int segEnd  = segBeg + segLen;
    const int rowBase = segBeg + blockIdx.y * BM;
    const int colBase = blockIdx.x * BN;          // gate column; up column = +INTER
    const int remRows = segEnd - rowBase;         // > 0 here; TDM zero-fills rows beyond

    const int tid  = threadIdx.x;
    const int lane = tid & 31;
    const int wave = tid >> 5;      // 0..7
    const int wm   = wave & 3;
    const int wn   = wave >> 2;
    const int l16  = lane & 15;
    const int half = lane >> 4;

    const float* wE     = w1w3 + (size_t)e * HIDDEN * (2 * INTER);
    const float* aTile  = x  + (size_t)rowBase * HIDDEN;
    const float* bgTile = wE + colBase;
    const float* buTile = wE + colBase + INTER;

    v8f accg0 = {}, accg1 = {}, accu0 = {}, accu1 = {};
    const int NT = HIDDEN / KC;
    int cur = 0;

#if USE_TDM
    auto issue = [&](int k0s, int buf) {
        tdm_load_tile(&As[buf][0], aTile + k0s,
                      KC, BM, (unsigned)(HIDDEN - k0s), (unsigned)remRows,
                      HIDDEN, 3u, 3u);
        tdm_load_tile(&Bg[buf][0], bgTile + (size_t)k0s * (2 * INTER),
                      BN, KC, (unsigned)(2 * INTER - colBase), (unsigned)(HIDDEN - k0s),
                      2 * INTER, 5u, 3u);
        tdm_load_tile(&Bu[buf][0], buTile + (size_t)k0s * (2 * INTER),
                      BN, KC, (unsigned)(INTER - colBase), (unsigned)(HIDDEN - k0s),
                      2 * INTER, 5u, 3u);
    };
    if (wave == 0) issue(0, 0);
#else
    const int arow = tid >> 2, acol = (tid & 3) * 4;
    const int brow = tid >> 4, bcol = (tid & 15) * 4;
    const bool arowOK = (rowBase + arow) < segEnd;
    const float* aSrc  = x + (size_t)(rowBase + arow) * HIDDEN + acol;
    const float* bgSrc = wE + (size_t)brow * (2 * INTER) + colBase + bcol;
    const float* buSrc = bgSrc + INTER;
#endif

    for (int kt = 0; kt < NT; ++kt) {
#if USE_TDM
        if (wave == 0) {
            if (kt + 1 < NT) {
                issue((kt + 1) * KC, cur ^ 1);
                __builtin_amdgcn_s_wait_tensorcnt(3);   // stage kt complete; kt+1 in flight
            } else {
                __builtin_amdgcn_s_wait_tensorcnt(0);
            }
        }
        __syncthreads();                                // publish stage kt to all waves
#else
        const int k0 = kt * KC;
        __syncthreads();
        float4 av = make_float4(0.f, 0.f, 0.f, 0.f);
        if (arowOK) av = *reinterpret_cast<const float4*>(aSrc + k0);
        const float4 bgv = *reinterpret_cast<const float4*>(bgSrc + (size_t)k0 * (2 * INTER));
        const float4 buv = *reinterpret_cast<const float4*>(buSrc + (size_t)k0 * (2 * INTER));
        *reinterpret_cast<float4*>(&As[cur][arow * LDA + acol]) = av;
        *reinterpret_cast<float4*>(&Bg[cur][brow * LDB + bcol]) = bgv;
        *reinterpret_cast<float4*>(&Bu[cur][brow * LDB + bcol]) = buv;
        __syncthreads();
#endif
        const float* Asc = &As[cur][0];
        const float* Bgc = &Bg[cur][0];
        const float* Buc = &Bu[cur][0];
        #pragma unroll
        for (int kk = 0; kk < KC; kk += 4) {
            const int kb = kk + half * 2;               // fp32 WMMA A/B K-pair layout
            const float2 af = *reinterpret_cast<const float2*>(&Asc[(wm * 16 + l16) * LDA + kb]);
            v2f a; a.x = af.x; a.y = af.y;
            const int c0 = wn * 32 + l16;
            v2f bg0, bg1, bu0, bu1;
            bg0.x = Bgc[kb * LDB + c0];        bg0.y = Bgc[(kb + 1) * LDB + c0];
            bg1.x = Bgc[kb * LDB + c0 + 16];   bg1.y = Bgc[(kb + 1) * LDB + c0 + 16];
            bu0.x = Buc[kb * LDB + c0];        bu0.y = Buc[(kb + 1) * LDB + c0];
            bu1.x = Buc[kb * LDB + c0 + 16];   bu1.y = Buc[(kb + 1) * LDB + c0 + 16];
            accg0 = wmma_f32(a, bg0, accg0);
            accg1 = wmma_f32(a, bg1, accg1);
            accu0 = wmma_f32(a, bu0, accu0);
            accu1 = wmma_f32(a, bu1, accu1);
        }
#if USE_TDM
        __syncthreads();                                // reads done before buffer reuse
#endif
        cur ^= 1;
    }

    // SwiGLU in registers (gate/up share the C/D lane layout), then store h
    #pragma unroll
    for (int i = 0; i < 8; ++i) {
        const int m    = wm * 16 + half * 8 + i;        // D: M = i + 8*(lane>=16)
        const int grow = rowBase + m;
        if (grow < segEnd) {
            const int c = colBase + wn * 32 + l16;      // D: N = lane&15
            const float g0 = accg0[i], u0 = accu0[i];
            const float g1 = accg1[i], u1 = accu1[i];
            const float s0 = g0 / (1.0f + __expf(-g0));
            const float s1 = g1 / (1.0f + __expf(-g1));
            h[(size_t)grow * INTER + c]      = s0 * u0;
            h[(size_t)grow * INTER + c + 16] = s1 * u1;
        }
    }
}

// ======================= Kernel 2: down projection =======================
__global__ __launch_bounds__(256) void moe_down_kernel(
    const float* __restrict__ h,       // [TOKENS, INTER]
    const float* __restrict__ w2,      // [NEXP, INTER, HIDDEN]
    const int*   __restrict__ tpe,     // [NEXP]
    float*       __restrict__ out)     // [TOKENS, HIDDEN]
{
    __shared__ float As[2][BM * LDA];
    __shared__ float Bs[2][KC * LDB];

    const int e      = blockIdx.z;
    const int segLen = __builtin_amdgcn_readfirstlane(tpe[e]);
    if ((int)blockIdx.y * BM >= segLen) return;
    const int segBeg  = __builtin_amdgcn_readfirstlane(seg_start(tpe, e));
    const int segEnd  = segBeg + segLen;
    const int rowBase = segBeg + blockIdx.y * BM;
    const int colBase = blockIdx.x * BN;
    const int remRows = segEnd - rowBase;

    const int tid  = threadIdx.x;
    const int lane = tid & 31;
    const int wave = tid >> 5;
    const int wm   = wave & 3;
    const int wn   = wave >> 2;
    const int l16  = lane & 15;
    const int half = lane >> 4;

    const float* wE    = w2 + (size_t)e * INTER * HIDDEN;
    const float* aTile = h  + (size_t)rowBase * INTER;
    const float* bTile = wE + colBase;

    v8f acc0 = {}, acc1 = {};
    const int NT = INTER / KC;   // 88
    int cur = 0;

#if USE_TDM
    auto issue = [&](int k0s, int buf) {
        tdm_load_tile(&As[buf][0], aTile + k0s,
                      KC, BM, (unsigned)(INTER - k0s), (unsigned)remRows,
                      INTER, 3u, 3u);
        tdm_load_tile(&Bs[buf][0], bTile + (size_t)k0s * HIDDEN,
                      BN, KC, (unsigned)(HIDDEN - colBase), (unsigned)(INTER - k0s),
                      HIDDEN, 5u, 3u);
    };
    if (wave == 0) issue(0, 0);
#else
    const int arow = tid >> 2, acol = (tid & 3) * 4;
    const int brow = tid >> 4, bcol = (tid & 15) * 4;
    const bool arowOK = (rowBase + arow) < segEnd;
    const float* aSrc = h  + (size_t)(rowBase + arow) * INTER + acol;
    const float* bSrc = wE + (size_t)brow * HIDDEN + colBase + bcol;
#endif

    for (int kt = 0; kt < NT; ++kt) {
#if USE_TDM
        if (wave == 0) {
            if (kt + 1 < NT) {
                issue((kt + 1) * KC, cur ^ 1);
                __builtin_amdgcn_s_wait_tensorcnt(2);
            } else {
                __builtin_amdgcn_s_wait_tensorcnt(0);
            }
        }
        __syncthreads();
#else
        const int k0 = kt * KC;
        __syncthreads();
        float4 av = make_float4(0.f, 0.f, 0.f, 0.f);
        if (arowOK) av = *reinterpret_cast<const float4*>(aSrc + k0);
        const float4 bv = *reinterpret_cast<const float4*>(bSrc + (size_t)k0 * HIDDEN);
        *reinterpret_cast<float4*>(&As[cur][arow * LDA + acol]) = av;
        *reinterpret_cast<float4*>(&Bs[cur][brow * LDB + bcol]) = bv;
        __syncthreads();
#endif
        const float* Asc = &As[cur][0];
        const float* Bsc = &Bs[cur][0];
        #pragma unroll
        for (int kk = 0; kk < KC; kk += 4) {
            const int kb = kk + half * 2;
            const float2 af = *reinterpret_cast<const float2*>(&Asc[(wm * 16 + l16) * LDA + kb]);
            v2f a; a.x = af.x; a.y = af.y;
            const int c0 = wn * 32 + l16;
            v2f b0, b1;
            b0.x = Bsc[kb * LDB + c0];        b0.y = Bsc[(kb + 1) * LDB + c0];
            b1.x = Bsc[kb * LDB + c0 + 16];   b1.y = Bsc[(kb + 1) * LDB + c0 + 16];
            acc0 = wmma_f32(a, b0, acc0);
            acc1 = wmma_f32(a, b1, acc1);
        }
#if USE_TDM
        __syncthreads();
#endif
        cur ^= 1;
    }

    #pragma unroll
    for (int i = 0; i < 8; ++i) {
        const int m    = wm * 16 + half * 8 + i;
        const int grow = rowBase + m;
        if (grow < segEnd) {
            const int c = colBase + wn * 32 + l16;
            out[(size_t)grow * HIDDEN + c]      = acc0[i];
            out[(size_t)grow * HIDDEN + c + 16] = acc1[i];
        }
    }
}

// ======================= Launch =======================
extern "C" void kernel_launch(void* const* d_in, const int* in_sizes, int n_in,
                              void* d_out, int out_size, void* d_ws, size_t ws_size,
                              hipStream_t stream) {
    (void)in_sizes; (void)n_in; (void)out_size; (void)ws_size;
    const float* x    = (const float*)d_in[0];   // [8192, 1024]
    const float* w1w3 = (const float*)d_in[1];   // [8, 1024, 2816]
    const float* w2   = (const float*)d_in[2];   // [8, 1408, 1024]
    const int*   tpe  = (const int*)d_in[3];     // [8]
    // d_in[4] = decoding flag (unused)
    float* out = (float*)d_out;                  // [8192, 1024]
    float* h   = (float*)d_ws;                   // [8192, 1408] fp32 scratch (46.1 MB)

    const int maxRowTiles = (TOKENS + BM - 1) / BM;
    dim3 blk(256, 1, 1);
    dim3 g1(INTER / BN,  maxRowTiles, NEXP);     // 22 x 128 x 8
    dim3 g2(HIDDEN / BN, maxRowTiles, NEXP);     // 16 x 128 x 8

    moe_gateup_swiglu_kernel<<<g1, blk, 0, stream>>>(x, w1w3, tpe, h);
    moe_down_kernel<<<g2, blk, 0, stream>>>(h, w2, tpe, out);
}